// DeepLabV3PlusMultiPrototypeSingleKeyDrop_56487409877272
// MI455X (gfx1250) — compile-verified
//
#include <hip/hip_runtime.h>

// ---------------------------------------------------------------------------
// DeepLabV3+ multi-prototype head, MI455X (gfx1250, wave32, WMMA).
//
// Math reductions (see analysis):
//  * softmax(sim,axis=1).mean(axis=1) == 1/12 exactly -> key/query branch dead.
//  * x = (assp'/12)·cls_w^T + cls_b where assp' only differs from assp at the
//    102 sampled (px,py) positions per batch -> fuse classify + proto-reduce +
//    x-store into ONE pass over assp, then fix up 102*16 positions, upsample.
// ---------------------------------------------------------------------------

typedef __attribute__((ext_vector_type(2))) float v2f;
typedef __attribute__((ext_vector_type(8))) float v8f;

#define BB   16
#define CC   6
#define DD   128
#define HH   128
#define WWI  128
#define HW   (HH * WWI)

// ---------------------------------------------------------------------------
// K0: zero the sums/counts workspace (ws is poisoned once, never re-zeroed).
// ---------------------------------------------------------------------------
__global__ void k_zero(float* __restrict__ p, int n) {
  int i = blockIdx.x * blockDim.x + threadIdx.x;
  if (i < n) p[i] = 0.0f;
}

// ---------------------------------------------------------------------------
// K1: fused classify + prototype accumulation + x-store.
// grid = (16 blocks per HW plane, 16 batches), block = 256 = 8 waves.
// Each wave: 8 tiles of 16 positions; per tile a 16x16x128 GEMM as 32
// v_wmma_f32_16x16x4_f32 ops (A = features, B = zero-padded cls_w^T).
// ---------------------------------------------------------------------------
__global__ __launch_bounds__(256) void k_classify(
    const float* __restrict__ assp, const float* __restrict__ cls_w,
    const float* __restrict__ cls_b, float* __restrict__ sums,
    float* __restrict__ counts, float* __restrict__ xbuf) {
  __shared__ float lsum[CC * DD];
  __shared__ float lcnt[CC];

  const int tid = threadIdx.x;
  const int b   = blockIdx.y;
  for (int i = tid; i < CC * DD; i += 256) lsum[i] = 0.0f;
  if (tid < CC) lcnt[tid] = 0.0f;
  __syncthreads();

  const int wv   = tid >> 5;
  const int lane = tid & 31;
  const int half = lane >> 4;   // K-half for A/B fragments, M-half for C/D
  const int pcol = lane & 15;   // A: row M (position) ; B/C/D: col N (class)

  // B fragment (cls_w^T, 128x16, cols >= 6 zero) held in VGPRs for all tiles.
  // Layout (16x16x4 f32): VGPR j of lane holds K = 4k + 2*half + j, N = pcol.
  v2f bfrag[32];
#pragma unroll
  for (int k = 0; k < 32; ++k) {
    const int d0 = 4 * k + 2 * half;
    if (pcol < CC) {
      bfrag[k].x = cls_w[pcol * DD + d0];
      bfrag[k].y = cls_w[pcol * DD + d0 + 1];
    } else {
      bfrag[k].x = 0.0f;
      bfrag[k].y = 0.0f;
    }
  }
  const float bias  = (pcol < CC) ? cls_b[pcol] : 0.0f;
  const float scale = 1.0f / 12.0f;  // simw == 1/12 (softmax-mean identity)

  const float* plane = assp + (size_t)b * DD * HW;

  for (int t = 0; t < 8; ++t) {
    const int tile    = blockIdx.x * 64 + wv * 8 + t;
    const int posBase = tile * 16;
    const float* fb   = plane + posBase + pcol;

    // A fragment: feature of position (posBase+pcol) at d = 4k + 2*half + j.
    // Kept in registers so the post-argmax LDS accumulation reuses it.
    v2f afrag[32];
    v8f acc = {};
#pragma unroll
    for (int k = 0; k < 32; ++k) {
      const int d0 = 4 * k + 2 * half;
      afrag[k].x = fb[(size_t)d0 * HW];
      afrag[k].y = fb[(size_t)(d0 + 1) * HW];
      acc = __builtin_amdgcn_wmma_f32_16x16x4_f32(
          false, afrag[k], false, bfrag[k], (short)0, acc, false, false);
    }
    __builtin_prefetch(fb + 16, 0, 0);  // next tile (global_prefetch_b8)

    // Store x = raw/12 + cls_b for this tile (positions posBase+8*half+i).
    if (pcol < CC) {
      float* xp = xbuf + ((size_t)(b * CC + pcol)) * HW + posBase + 8 * half;
      float4 lo = make_float4(acc[0] * scale + bias, acc[1] * scale + bias,
                              acc[2] * scale + bias, acc[3] * scale + bias);
      float4 hi = make_float4(acc[4] * scale + bias, acc[5] * scale + bias,
                              acc[6] * scale + bias, acc[7] * scale + bias);
      *reinterpret_cast<float4*>(xp)     = lo;
      *reinterpret_cast<float4*>(xp + 4) = hi;
    }

    // Cross-lane argmax over classes (lanes 0..5 of each 16-lane half),
    // first-index-wins on ties to match jnp.argmax.
    unsigned packed = 0u;
#pragma unroll
    for (int i = 0; i < 8; ++i) {
      float v  = (pcol < CC) ? (acc[i] + bias) : -3.402823e38f;
      int  idx = pcol;
#pragma unroll
      for (int m = 1; m <= 8; m <<= 1) {
        float ov = __shfl_xor(v, m, 32);
        int   oi = __shfl_xor(idx, m, 32);
        if (ov > v || (ov == v && oi < idx)) { v = ov; idx = oi; }
      }
      packed |= (unsigned)idx << (4 * i);  // class of position i + 8*half
    }
    // Exchange packed class lists between halves; pick class of position pcol.
    const unsigned other = __shfl_xor(packed, 16, 32);
    const int hp   = pcol >> 3;
    const unsigned srcp = (hp == half) ? packed : other;
    const int myCls = (int)((srcp >> (4 * (pcol & 7))) & 0xFu);

    if (half == 0) atomicAdd(&lcnt[myCls], 1.0f);  // each position once
#pragma unroll
    for (int k = 0; k < 32; ++k) {
      const int d0 = 4 * k + 2 * half;
      atomicAdd(&lsum[myCls * DD + d0],     afrag[k].x);
      atomicAdd(&lsum[myCls * DD + d0 + 1], afrag[k].y);
    }
  }

  __syncthreads();
  for (int i = tid; i < CC * DD; i += 256)
    atomicAdd(&sums[b * CC * DD + i], lsum[i]);
  if (tid < CC) atomicAdd(&counts[b * CC + tid], lcnt[tid]);
}

// ---------------------------------------------------------------------------
// K2: finalize prototypes, nearest-prototype lookup at sampled positions,
// and overwrite x there with (proto·cls_w^T)/12 + cls_b.  grid = B blocks.
// (protos12 duplicates each class twice -> argmin over 12 == argmin over 6.)
// ---------------------------------------------------------------------------
__global__ __launch_bounds__(256) void k_protos_fixup(
    const float* __restrict__ assp, const float* __restrict__ cls_w,
    const float* __restrict__ cls_b, const int* __restrict__ px,
    const int* __restrict__ py, int P, const float* __restrict__ sums,
    const float* __restrict__ counts, float* __restrict__ xbuf) {
  __shared__ float proto[CC * DD];
  const int tid = threadIdx.x;
  const int b   = blockIdx.x;

  for (int i = tid; i < CC * DD; i += 256) {
    const int   cl  = i >> 7;
    const float cnt = counts[b * CC + cl];
    const float s   = sums[b * CC * DD + i];
    proto[i] = (cnt > 0.0f) ? s / (cnt + 1e-5f) : 0.0f;
  }
  __syncthreads();

  const int wv   = tid >> 5;
  const int lane = tid & 31;
  const int d4   = lane * 4;

  for (int p = wv; p < P; p += 8) {
    const int h = px[p], w = py[p];
    const size_t pos = (size_t)h * WWI + w;
    const float* fb  = assp + (size_t)b * DD * HW + pos;
    float f0 = fb[(size_t)(d4 + 0) * HW];
    float f1 = fb[(size_t)(d4 + 1) * HW];
    float f2 = fb[(size_t)(d4 + 2) * HW];
    float f3 = fb[(size_t)(d4 + 3) * HW];

    float best = 3.402823e38f;
    int   bk   = 0;
#pragma unroll
    for (int cl = 0; cl < CC; ++cl) {
      const float* pr = proto + cl * DD + d4;
      float g0 = pr[0] - f0, g1 = pr[1] - f1, g2 = pr[2] - f2, g3 = pr[3] - f3;
      float part = g0 * g0 + g1 * g1 + g2 * g2 + g3 * g3;
#pragma unroll
      for (int m = 1; m < 32; m <<= 1) part += __shfl_xor(part, m, 32);
      if (part < best) { best = part; bk = cl; }  // uniform across wave
    }

#pragma unroll
    for (int co = 0; co < CC; ++co) {
      const float* pr = proto + bk * DD + d4;
      const float* cw = cls_w + co * DD + d4;
      float part = pr[0] * cw[0] + pr[1] * cw[1] + pr[2] * cw[2] + pr[3] * cw[3];
#pragma unroll
      for (int m = 1; m < 32; m <<= 1) part += __shfl_xor(part, m, 32);
      if (lane == 0)
        xbuf[((size_t)(b * CC + co)) * HW + pos] =
            part * (1.0f / 12.0f) + cls_b[co];
    }
  }
}

// ---------------------------------------------------------------------------
// K3: separable 4x bilinear upsample (align-corners style, weights in f64 to
// match numpy's _lin_mat bit-for-bit).  grid = (512 rows, B*C planes).
// ---------------------------------------------------------------------------
__global__ __launch_bounds__(256) void k_upsample(
    const float* __restrict__ xbuf, float* __restrict__ out) {
  __shared__ float r0[WWI];
  __shared__ float r1[WWI];
  const int Y     = blockIdx.x;
  const int plane = blockIdx.y;
  const int tid   = threadIdx.x;

  const double sy = (double)Y * 127.0 / 511.0;
  const int   ly  = (int)sy;
  const float wy  = (float)(sy - (double)ly);
  const int   hy  = (ly + 1 < 127) ? ly + 1 : 127;

  const float* xp = xbuf + (size_t)plane * HW;
  if (tid < 128) {
    r0[tid] = xp[ly * WWI + tid];
  } else {
    r1[tid - 128] = xp[hy * WWI + (tid - 128)];
  }
  __syncthreads();

  for (int X = tid; X < 512; X += 256) {
    const double sx = (double)X * 127.0 / 511.0;
    const int   lx  = (int)sx;
    const float wx  = (float)(sx - (double)lx);
    const int   hx  = (lx + 1 < 127) ? lx + 1 : 127;
    const float top = (1.0f - wx) * r0[lx] + wx * r0[hx];
    const float bot = (1.0f - wx) * r1[lx] + wx * r1[hx];
    out[((size_t)plane * 512 + Y) * 512 + X] = (1.0f - wy) * top + wy * bot;
  }
}

// ---------------------------------------------------------------------------
extern "C" void kernel_launch(void* const* d_in, const int* in_sizes, int n_in,
                              void* d_out, int out_size, void* d_ws,
                              size_t ws_size, hipStream_t stream) {
  const float* assp  = (const float*)d_in[0];
  const float* cls_w = (const float*)d_in[1];
  const float* cls_b = (const float*)d_in[2];
  // d_in[3..6] (key_w,key_b,query_w,query_b) mathematically dead: the
  // softmax-mean over the 12 similarity channels is identically 1/12.
  const int* px = (const int*)d_in[7];
  const int* py = (const int*)d_in[8];
  const int  P  = in_sizes[7];

  float* ws     = (float*)d_ws;
  float* sums   = ws;             // B*C*D = 12288 floats
  float* counts = ws + 12288;     // B*C   = 96 floats
  float* xbuf   = ws + 12384;     // B*C*HW = 1,572,864 floats (16B aligned)
  (void)n_in; (void)ws_size; (void)out_size;

  const int nz = BB * CC * DD + BB * CC;
  k_zero<<<(nz + 255) / 256, 256, 0, stream>>>(ws, nz);

  k_classify<<<dim3(16, BB), 256, 0, stream>>>(assp, cls_w, cls_b, sums,
                                               counts, xbuf);

  k_protos_fixup<<<BB, 256, 0, stream>>>(assp, cls_w, cls_b, px, py, P, sums,
                                         counts, xbuf);

  k_upsample<<<dim3(512, BB * CC), 256, 0, stream>>>(xbuf, (float*)d_out);
}